// Model_NodeEdge_50869592655527
// MI455X (gfx1250) — compile-verified
//
#include <hip/hip_runtime.h>

typedef __attribute__((ext_vector_type(2))) float v2f;
typedef __attribute__((ext_vector_type(8))) float v8f;

// ---------------------------------------------------------------------------
// WMMA GEMM:  Y[M,N] = act(A)[M,K] @ B[K,N]   (B row-major [K][N])
// If FUSE: act(x) = relu(x*inScale[k] + inShift[k]) applied to A on load.
// Epilogue: accumulate column sums / sums-of-squares into stats[0:N] / stats[N:2N].
// Block = 256 threads = 8 waves; each wave computes 16 rows x N cols.
// B is staged into LDS pre-swizzled as {B[2p][col], B[2p+1][col]} float2 pairs so
// each WMMA B-fragment is one aligned ds_load_b64 into an even VGPR pair.
// ---------------------------------------------------------------------------
template <int K, int N, bool FUSE>
__global__ __launch_bounds__(256) void k_gemm(const float* __restrict__ A,
                                              const float* __restrict__ B,
                                              float* __restrict__ Y, int M,
                                              const float* __restrict__ inScale,
                                              const float* __restrict__ inShift,
                                              float* __restrict__ stats) {
  constexpr int NT = N / 16;
  constexpr int KC = 64;
  __shared__ float2 bswz[(KC / 2) * N];  // [k-pair p][col]
  __shared__ float s_sum[N];
  __shared__ float s_sq[N];
  __shared__ float s_scale[FUSE ? K : 1];
  __shared__ float s_shift[FUSE ? K : 1];

  const int tid = threadIdx.x;
  const int wave = tid >> 5;
  const int lane = tid & 31;
  const int lhalf = lane & 15;
  const int sel = (lane >> 4) << 1;  // 0 for lanes 0-15, 2 for lanes 16-31
  const int rowA = blockIdx.x * 128 + wave * 16 + lhalf;
  // unconditional loads from a clamped row + 0/1 mask (only last block has pads;
  // masked rows contribute exact zeros to the BN statistics)
  const float amask = (rowA < M) ? 1.f : 0.f;
  const float* __restrict__ Arow = A + (size_t)(rowA < M ? rowA : 0) * K;

  if (tid < N) { s_sum[tid] = 0.f; s_sq[tid] = 0.f; }
  if constexpr (FUSE) {
    for (int i = tid; i < K; i += 256) { s_scale[i] = inScale[i]; s_shift[i] = inShift[i]; }
  }

  v8f acc[NT];
#pragma unroll
  for (int nt = 0; nt < NT; ++nt)
#pragma unroll
    for (int v = 0; v < 8; ++v) acc[nt][v] = 0.f;

  for (int kc = 0; kc < K; kc += KC) {
    __syncthreads();
    // stage B chunk pre-swizzled (strided global reads are coalesced along col)
    for (int i = tid; i < (KC / 2) * N; i += 256) {
      const int p = i / N, col = i - p * N;
      bswz[i] = make_float2(B[(size_t)(kc + 2 * p) * N + col],
                            B[(size_t)(kc + 2 * p + 1) * N + col]);
    }
    if (kc + KC < K) __builtin_prefetch(B + (size_t)(kc + KC) * N + tid * 16, 0, 1);
    __syncthreads();

#pragma unroll 4
    for (int kk = 0; kk < KC; kk += 4) {
      const int kg = kc + kk + sel;
      float2 t = *(const float2*)(Arow + kg);
      if constexpr (FUSE) {
        t.x = fmaxf(fmaf(t.x, s_scale[kg], s_shift[kg]), 0.f);
        t.y = fmaxf(fmaf(t.y, s_scale[kg + 1], s_shift[kg + 1]), 0.f);
      }
      v2f a;
      a[0] = t.x * amask;
      a[1] = t.y * amask;
      const int p = (kk + sel) >> 1;
#pragma unroll
      for (int nt = 0; nt < NT; ++nt) {
        v2f b = *(const v2f*)&bswz[p * N + nt * 16 + lhalf];
        acc[nt] = __builtin_amdgcn_wmma_f32_16x16x4_f32(false, a, false, b, (short)0,
                                                        acc[nt], false, false);
      }
    }
  }

  // store + column stats (padded rows contribute exact zeros)
  const int rowBase = blockIdx.x * 128 + wave * 16 + ((lane >> 4) << 3);
#pragma unroll
  for (int nt = 0; nt < NT; ++nt) {
    float s = 0.f, q = 0.f;
#pragma unroll
    for (int v = 0; v < 8; ++v) {
      const int row = rowBase + v;
      const float y = acc[nt][v];
      if (row < M) Y[(size_t)row * N + nt * 16 + lhalf] = y;
      s += y;
      q += y * y;
    }
    s += __shfl_down(s, 16, 32);
    q += __shfl_down(q, 16, 32);
    if (lane < 16) {
      atomicAdd(&s_sum[nt * 16 + lane], s);
      atomicAdd(&s_sq[nt * 16 + lane], q);
    }
  }
  __syncthreads();
  if (tid < N) {
    atomicAdd(&stats[tid], s_sum[tid]);
    atomicAdd(&stats[N + tid], s_sq[tid]);
  }
}

// ---------------------------------------------------------------------------
// Elementwise / scatter kernels
// ---------------------------------------------------------------------------
__global__ void k_zero(float* p, int n) {
  int i = blockIdx.x * blockDim.x + threadIdx.x;
  if (i < n) p[i] = 0.f;
}

__global__ void k_node_embed(const int* __restrict__ node_x, const float* __restrict__ emb,
                             float* __restrict__ nodeRep, int n16) {
  int i = blockIdx.x * blockDim.x + threadIdx.x;
  if (i >= n16) return;
  int node = i >> 4, c = i & 15;
  ((float4*)nodeRep)[node * 16 + c] = ((const float4*)emb)[node_x[node] * 16 + c];
}

__global__ void k_edge_scatter_init(const int* __restrict__ edge_attr,
                                    const int* __restrict__ src, const int* __restrict__ dst,
                                    const float* __restrict__ eemb, float* __restrict__ acc,
                                    int e16) {
  int i = blockIdx.x * blockDim.x + threadIdx.x;
  if (i >= e16) return;
  int e = i >> 4, c = i & 15;
  float4 v = ((const float4*)eemb)[edge_attr[e] * 16 + c];
  float* ps = acc + (size_t)src[e] * 64 + c * 4;
  float* pd = acc + (size_t)dst[e] * 64 + c * 4;
  atomicAdd(ps + 0, v.x); atomicAdd(ps + 1, v.y); atomicAdd(ps + 2, v.z); atomicAdd(ps + 3, v.w);
  atomicAdd(pd + 0, v.x); atomicAdd(pd + 1, v.y); atomicAdd(pd + 2, v.z); atomicAdd(pd + 3, v.w);
}

__global__ void k_gather_pairs(const int* __restrict__ src, const int* __restrict__ dst,
                               const float* __restrict__ acc, float* __restrict__ edgeRep,
                               int e16) {
  int i = blockIdx.x * blockDim.x + threadIdx.x;
  if (i >= e16) return;
  int e = i >> 4, c = i & 15;
  ((float4*)edgeRep)[(size_t)(e * 2 + 0) * 16 + c] = ((const float4*)acc)[(size_t)src[e] * 16 + c];
  ((float4*)edgeRep)[(size_t)(e * 2 + 1) * 16 + c] = ((const float4*)acc)[(size_t)dst[e] * 16 + c];
}

__global__ void k_build_h(const int* __restrict__ src, const int* __restrict__ dst,
                          const float* __restrict__ edgeRep, const float* __restrict__ nodeRep,
                          float* __restrict__ h, int e16) {
  int i = blockIdx.x * blockDim.x + threadIdx.x;
  if (i >= e16) return;
  int e = i >> 4, c = i & 15;
  float4 ns = ((const float4*)nodeRep)[(size_t)src[e] * 16 + c];
  float4 nd = ((const float4*)nodeRep)[(size_t)dst[e] * 16 + c];
  float4 sm = make_float4(ns.x + nd.x, ns.y + nd.y, ns.z + nd.z, ns.w + nd.w);
  float4* H = (float4*)h;
  size_t r0 = (size_t)(e * 2) * 48, r1 = r0 + 48;  // 192 floats = 48 float4 per row
  H[r0 + c] = ((const float4*)edgeRep)[(size_t)(e * 2) * 16 + c];
  H[r0 + 16 + c] = ns;
  H[r0 + 32 + c] = sm;
  H[r1 + c] = ((const float4*)edgeRep)[(size_t)(e * 2 + 1) * 16 + c];
  H[r1 + 16 + c] = nd;
  H[r1 + 32 + c] = sm;
}

__global__ void k_prep_affine(const float* __restrict__ stats, const float* __restrict__ g,
                              const float* __restrict__ b, float* __restrict__ scale,
                              float* __restrict__ shift, int ncols, float invM) {
  int j = blockIdx.x * blockDim.x + threadIdx.x;
  if (j >= ncols) return;
  float m = stats[j] * invM;
  float var = stats[ncols + j] * invM - m * m;
  float sc = g[j] * rsqrtf(var + 1e-5f);
  scale[j] = sc;
  shift[j] = b[j] - m * sc;
}

// in-place BN+ReLU on [2E,64] rows, then scatter-add into e2n
__global__ void k_finalize_edge(float* __restrict__ Y, const float* __restrict__ scale,
                                const float* __restrict__ shift, const int* __restrict__ src,
                                const int* __restrict__ dst, float* __restrict__ e2n, int r16) {
  int i = blockIdx.x * blockDim.x + threadIdx.x;
  if (i >= r16) return;
  int row = i >> 4, c = i & 15;
  int e = row >> 1, j = row & 1;
  int col = c * 4;
  float4 y = ((float4*)Y)[(size_t)row * 16 + c];
  y.x = fmaxf(fmaf(y.x, scale[col + 0], shift[col + 0]), 0.f);
  y.y = fmaxf(fmaf(y.y, scale[col + 1], shift[col + 1]), 0.f);
  y.z = fmaxf(fmaf(y.z, scale[col + 2], shift[col + 2]), 0.f);
  y.w = fmaxf(fmaf(y.w, scale[col + 3], shift[col + 3]), 0.f);
  ((float4*)Y)[(size_t)row * 16 + c] = y;
  int node = j ? dst[e] : src[e];
  float* p = e2n + (size_t)node * 64 + col;
  atomicAdd(p + 0, y.x); atomicAdd(p + 1, y.y); atomicAdd(p + 2, y.z); atomicAdd(p + 3, y.w);
}

__global__ void k_build_xn(const float* __restrict__ nodeRep, const float* __restrict__ e2n,
                           float* __restrict__ Xn, int n32) {
  int i = blockIdx.x * blockDim.x + threadIdx.x;
  if (i >= n32) return;
  int n = i >> 5, c = i & 31;
  ((float4*)Xn)[(size_t)n * 32 + c] =
      (c < 16) ? ((const float4*)nodeRep)[(size_t)n * 16 + c]
               : ((const float4*)e2n)[(size_t)n * 16 + (c - 16)];
}

// BN+ReLU on [rows,64] from Y into out
__global__ void k_affine_relu64(const float* __restrict__ Y, const float* __restrict__ scale,
                                const float* __restrict__ shift, float* __restrict__ outb,
                                int r16) {
  int i = blockIdx.x * blockDim.x + threadIdx.x;
  if (i >= r16) return;
  int c = i & 15, col = c * 4;
  float4 y = ((const float4*)Y)[i];
  y.x = fmaxf(fmaf(y.x, scale[col + 0], shift[col + 0]), 0.f);
  y.y = fmaxf(fmaf(y.y, scale[col + 1], shift[col + 1]), 0.f);
  y.z = fmaxf(fmaf(y.z, scale[col + 2], shift[col + 2]), 0.f);
  y.w = fmaxf(fmaf(y.w, scale[col + 3], shift[col + 3]), 0.f);
  ((float4*)outb)[i] = y;
}

__global__ void k_pool_scatter(const int* __restrict__ batch, const float* __restrict__ nodeRep,
                               const float* __restrict__ e2n, float* __restrict__ psum, int n32) {
  int i = blockIdx.x * blockDim.x + threadIdx.x;
  if (i >= n32) return;
  int n = i >> 5, c = i & 31;
  float4 v = (c < 16) ? ((const float4*)nodeRep)[(size_t)n * 16 + c]
                      : ((const float4*)e2n)[(size_t)n * 16 + (c - 16)];
  float* p = psum + (size_t)batch[n] * 128 + c * 4;
  atomicAdd(p + 0, v.x); atomicAdd(p + 1, v.y); atomicAdd(p + 2, v.z); atomicAdd(p + 3, v.w);
}

__global__ void k_count(const int* __restrict__ batch, float* __restrict__ cnt, int n) {
  int i = blockIdx.x * blockDim.x + threadIdx.x;
  if (i < n) atomicAdd(&cnt[batch[i]], 1.0f);
}

__global__ void k_pool_div(const float* __restrict__ psum, const float* __restrict__ cnt,
                           float* __restrict__ pooled, int total) {
  int i = blockIdx.x * blockDim.x + threadIdx.x;
  if (i >= total) return;
  pooled[i] = psum[i] / fmaxf(cnt[i >> 7], 1.f);
}

__global__ void k_head_final(const float* __restrict__ Y, const float* __restrict__ scale,
                             const float* __restrict__ shift, const float* __restrict__ linW,
                             const float* __restrict__ linb, float* __restrict__ out) {
  int g = threadIdx.x;  // 256 graphs, 1 block
  float a = 0.f;
  for (int j = 0; j < 128; ++j) {
    float x = fmaxf(fmaf(Y[(size_t)g * 128 + j], scale[j], shift[j]), 0.f);
    a = fmaf(x, linW[j], a);
  }
  out[g] = a + linb[0];
}

// ---------------------------------------------------------------------------
extern "C" void kernel_launch(void* const* d_in, const int* in_sizes, int n_in,
                              void* d_out, int out_size, void* d_ws, size_t ws_size,
                              hipStream_t stream) {
  constexpr int NN = 25000, EE = 250000, NG = 256, L = 3;
  const int M2E = 2 * EE;

  const int* node_x = (const int*)d_in[0];
  const int* edge_attr = (const int*)d_in[1];
  const int* srcI = (const int*)d_in[2];
  const int* dstI = srcI + EE;
  const int* batch = (const int*)d_in[3];
  const float* node_emb = (const float*)d_in[4];
  const float* edge_emb = (const float*)d_in[5];
  const float* eW1 = (const float*)d_in[6];
  const float* eg1 = (const float*)d_in[7];
  const float* eb1 = (const float*)d_in[8];
  const float* eW2 = (const float*)d_in[9];
  const float* eg2 = (const float*)d_in[10];
  const float* eb2 = (const float*)d_in[11];
  const float* nW1 = (const float*)d_in[12];
  const float* ng1 = (const float*)d_in[13];
  const float* nb1 = (const float*)d_in[14];
  const float* nW2 = (const float*)d_in[15];
  const float* ng2 = (const float*)d_in[16];
  const float* nb2 = (const float*)d_in[17];
  const float* fW1 = (const float*)d_in[18];
  const float* fg1 = (const float*)d_in[19];
  const float* fb1 = (const float*)d_in[20];
  const float* fW2 = (const float*)d_in[21];
  const float* fg2 = (const float*)d_in[22];
  const float* fb2 = (const float*)d_in[23];
  const float* linW = (const float*)d_in[24];
  const float* linb = (const float*)d_in[25];
  float* out = (float*)d_out;

  // workspace carve (256B aligned)
  char* wp = (char*)d_ws;
  auto alloc = [&](size_t nf) {
    float* r = (float*)wp;
    wp += (nf * sizeof(float) + 255) & ~((size_t)255);
    return r;
  };
  float* nodeRep = alloc((size_t)NN * 64);
  float* e2n = alloc((size_t)NN * 64);       // also initial edge-accumulator
  float* edgeRep = alloc((size_t)M2E * 64);  // also Y2 of edge MLP (in-place BN)
  float* hbuf = alloc((size_t)M2E * 192);
  float* Y1 = alloc((size_t)M2E * 128);
  float* Xn = alloc((size_t)NN * 128);
  float* Yn1 = alloc((size_t)NN * 128);
  float* Yn2 = alloc((size_t)NN * 64);
  float* stats = alloc(256);
  float* sc1 = alloc(128);
  float* sh1 = alloc(128);
  float* sc2 = alloc(128);
  float* sh2 = alloc(128);
  float* psum = alloc((size_t)NG * 128);
  float* cnt = alloc(NG);
  float* pooled = alloc((size_t)NG * 128);
  float* Yh1 = alloc((size_t)NG * 128);
  float* Yh2 = alloc((size_t)NG * 128);

  auto blocks = [](long long n) { return (unsigned)((n + 255) / 256); };
  const unsigned gEdge = (unsigned)((M2E + 127) / 128);
  const unsigned gNode = (unsigned)((NN + 127) / 128);
  const unsigned gHead = (unsigned)((NG + 127) / 128);

  // ---- embeddings + initial edge-incidence accumulation ----
  k_zero<<<blocks((long long)NN * 64), 256, 0, stream>>>(e2n, NN * 64);
  k_node_embed<<<blocks((long long)NN * 16), 256, 0, stream>>>(node_x, node_emb, nodeRep, NN * 16);
  k_edge_scatter_init<<<blocks((long long)EE * 16), 256, 0, stream>>>(edge_attr, srcI, dstI,
                                                                      edge_emb, e2n, EE * 16);
  k_gather_pairs<<<blocks((long long)EE * 16), 256, 0, stream>>>(srcI, dstI, e2n, edgeRep,
                                                                 EE * 16);

  for (int l = 0; l < L; ++l) {
    const float* eW1l = eW1 + (size_t)l * 192 * 128;
    const float* eW2l = eW2 + (size_t)l * 128 * 64;
    const float* nW1l = nW1 + (size_t)l * 128 * 128;
    const float* nW2l = nW2 + (size_t)l * 128 * 64;

    // ---- edge MLP ----
    k_build_h<<<blocks((long long)EE * 16), 256, 0, stream>>>(srcI, dstI, edgeRep, nodeRep, hbuf,
                                                              EE * 16);
    k_zero<<<1, 256, 0, stream>>>(stats, 256);
    k_gemm<192, 128, false><<<gEdge, 256, 0, stream>>>(hbuf, eW1l, Y1, M2E, nullptr, nullptr,
                                                       stats);
    k_prep_affine<<<1, 128, 0, stream>>>(stats, eg1 + l * 128, eb1 + l * 128, sc1, sh1, 128,
                                         1.0f / M2E);
    k_zero<<<1, 256, 0, stream>>>(stats, 256);
    k_gemm<128, 64, true><<<gEdge, 256, 0, stream>>>(Y1, eW2l, edgeRep, M2E, sc1, sh1, stats);
    k_prep_affine<<<1, 64, 0, stream>>>(stats, eg2 + l * 64, eb2 + l * 64, sc2, sh2, 64,
                                        1.0f / M2E);
    k_zero<<<blocks((long long)NN * 64), 256, 0, stream>>>(e2n, NN * 64);
    k_finalize_edge<<<blocks((long long)M2E * 16), 256, 0, stream>>>(edgeRep, sc2, sh2, srcI,
                                                                     dstI, e2n, M2E * 16);

    // ---- node MLP ----
    k_build_xn<<<blocks((long long)NN * 32), 256, 0, stream>>>(nodeRep, e2n, Xn, NN * 32);
    k_zero<<<1, 256, 0, stream>>>(stats, 256);
    k_gemm<128, 128, false><<<gNode, 256, 0, stream>>>(Xn, nW1l, Yn1, NN, nullptr, nullptr,
                                                       stats);
    k_prep_affine<<<1, 128, 0, stream>>>(stats, ng1 + l * 128, nb1 + l * 128, sc1, sh1, 128,
                                         1.0f / NN);
    k_zero<<<1, 256, 0, stream>>>(stats, 256);
    k_gemm<128, 64, true><<<gNode, 256, 0, stream>>>(Yn1, nW2l, Yn2, NN, sc1, sh1, stats);
    k_prep_affine<<<1, 64, 0, stream>>>(stats, ng2 + l * 64, nb2 + l * 64, sc2, sh2, 64,
                                        1.0f / NN);
    k_affine_relu64<<<blocks((long long)NN * 16), 256, 0, stream>>>(Yn2, sc2, sh2, nodeRep,
                                                                    NN * 16);
  }

  // ---- readout (post-loop e2n == last layer's e2n, reuse it) ----
  k_zero<<<blocks((long long)NG * 128), 256, 0, stream>>>(psum, NG * 128);
  k_zero<<<1, 256, 0, stream>>>(cnt, NG);
  k_pool_scatter<<<blocks((long long)NN * 32), 256, 0, stream>>>(batch, nodeRep, e2n, psum,
                                                                 NN * 32);
  k_count<<<blocks(NN), 256, 0, stream>>>(batch, cnt, NN);
  k_pool_div<<<blocks((long long)NG * 128), 256, 0, stream>>>(psum, cnt, pooled, NG * 128);

  // ---- head MLP ----
  k_zero<<<1, 256, 0, stream>>>(stats, 256);
  k_gemm<128, 128, false><<<gHead, 256, 0, stream>>>(pooled, fW1, Yh1, NG, nullptr, nullptr,
                                                     stats);
  k_prep_affine<<<1, 128, 0, stream>>>(stats, fg1, fb1, sc1, sh1, 128, 1.0f / NG);
  k_zero<<<1, 256, 0, stream>>>(stats, 256);
  k_gemm<128, 128, true><<<gHead, 256, 0, stream>>>(Yh1, fW2, Yh2, NG, sc1, sh1, stats);
  k_prep_affine<<<1, 128, 0, stream>>>(stats, fg2, fb2, sc2, sh2, 128, 1.0f / NG);
  k_head_final<<<1, 256, 0, stream>>>(Yh2, sc2, sh2, linW, linb, out);

  (void)in_sizes; (void)n_in; (void)out_size; (void)ws_size;
}